// SACRSN_49417893707907
// MI455X (gfx1250) — compile-verified
//
#include <hip/hip_runtime.h>
#include <hip/hip_bf16.h>
#include <math.h>

#define DD 1024
#define DD2 2048
#define NSYM 8192
#define VOCAB 50257
#define STACKN 64
#define EPSF 1e-6f
#define ACT_TF 0.999f
#define GBSF 0.5f
#define CCF 0.25f

typedef __attribute__((ext_vector_type(2))) float v2f;
typedef __attribute__((ext_vector_type(8))) float v8f;

// ---- workspace layout (float offsets) ----
#define OFF_ZR      0
#define OFF_ZI      1024
#define OFF_RR      2048
#define OFF_II      3072
#define OFF_ZF      4096          // 2048
#define OFF_SCAL    6144          // 32 scalars: 0=p 1=push 2=pop 3=noop 4=halt 5=remain 6=ponder 7=vq_loss
#define OFF_SYM     6176          // int slot
#define OFF_PTR     6208          // 64
#define OFF_ZRACC   6272          // 1024
#define OFF_ZIACC   7296          // 1024
#define OFF_MEM     8320          // 64*1024
#define OFF_CBN     73856         // 8192
#define OFF_DOTS    82048         // 50272 (padded)
// end ~132320 floats (~529 KB)

static __device__ inline v8f wmma4(v2f a, v2f b, v8f c) {
  return __builtin_amdgcn_wmma_f32_16x16x4_f32(false, a, false, b, (short)0, c, false, false);
}

static __device__ inline float block_reduce_sum(float v, float* sbuf) {
  int t = threadIdx.x;
  sbuf[t] = v;
  __syncthreads();
  for (int s = blockDim.x >> 1; s > 0; s >>= 1) {
    if (t < s) sbuf[t] += sbuf[t + s];
    __syncthreads();
  }
  float r = sbuf[0];
  __syncthreads();
  return r;
}

// ---------- init: embeddings -> (zr,zi), zero state ----------
__global__ void k_init(const int* x, const float* mag_emb, const float* phase_emb, float* ws) {
  int gid = blockIdx.x * blockDim.x + threadIdx.x;
  if (gid < STACKN * DD) ws[OFF_MEM + gid] = 0.0f;
  if (gid < DD) {
    int idx = x[0];
    float m = mag_emb[(long)idx * DD + gid];
    float ph = phase_emb[(long)idx * DD + gid];
    ws[OFF_ZR + gid] = m * cosf(ph);
    ws[OFF_ZI + gid] = m * sinf(ph);
    ws[OFF_ZRACC + gid] = 0.0f;
    ws[OFF_ZIACC + gid] = 0.0f;
  }
  if (gid < STACKN) ws[OFF_PTR + gid] = (gid == 0) ? 1.0f : 0.0f;
  if (gid == 0) {
    for (int i = 0; i < 32; ++i) ws[OFF_SCAL + i] = 0.0f;
    ws[OFF_SCAL + 5] = 1.0f;  // remain
  }
}

// ---------- codebook row squared norms (one block per row) ----------
__global__ void k_cbnorm(const float* codebook, float* ws) {
  __shared__ float sbuf[256];
  int row = blockIdx.x;
  const float* cr = codebook + (long)row * DD2;
  float acc = 0.0f;
  for (int k = threadIdx.x; k < DD2; k += 256) { float v = cr[k]; acc += v * v; }
  float s = block_reduce_sum(acc, sbuf);
  if (threadIdx.x == 0) ws[OFF_CBN + row] = s;
}

// ---------- fused complex matvec: rr = Wr z_r - Wi z_i ; ii = Wr z_i + Wi z_r ----------
// one wave (32 threads) per 16 output rows; WMMA f32 16x16x4
__global__ void k_cplx_mv(const float* Wr, const float* Wi, const float* ws_in, float* ws) {
  int lane = threadIdx.x;
  int m = lane & 15, half = lane >> 4;
  int rbase = blockIdx.x * 16;
  const float* wr_row = Wr + (long)(rbase + m) * DD;
  const float* wi_row = Wi + (long)(rbase + m) * DD;
  const float* zr = ws_in + OFF_ZR;
  const float* zi = ws_in + OFF_ZI;
  v8f acc_rr = {};
  v8f acc_ii = {};
#pragma unroll 4
  for (int k = 0; k < DD; k += 4) {
    int kk = k + half * 2;
    v2f awr = *(const v2f*)(wr_row + kk);
    v2f awi = *(const v2f*)(wi_row + kk);
    v2f bzr = *(const v2f*)(zr + kk);
    v2f bzi = *(const v2f*)(zi + kk);
    v2f bnzi = -bzi;
    acc_rr = wmma4(awr, bzr, acc_rr);
    acc_rr = wmma4(awi, bnzi, acc_rr);
    acc_ii = wmma4(awr, bzi, acc_ii);
    acc_ii = wmma4(awi, bzr, acc_ii);
  }
  if (m == 0) {
    int ob = rbase + half * 8;
    for (int v = 0; v < 8; ++v) {
      ws[OFF_RR + ob + v] = acc_rr[v];
      ws[OFF_II + ob + v] = acc_ii[v];
    }
  }
}

// ---------- generic matvec: out[r] = dot(M[r,:], vec) via WMMA ----------
__global__ void k_mv_dot(const float* M, const float* vec, float* out, int rows, int K) {
  int lane = threadIdx.x;
  int m = lane & 15, half = lane >> 4;
  int rbase = blockIdx.x * 16;
  int row = rbase + m;
  if (row >= rows) row = rows - 1;  // clamp loads, guard stores
  const float* Mr = M + (long)row * K;
  v8f acc = {};
#pragma unroll 4
  for (int k = 0; k < K; k += 4) {
    int kk = k + half * 2;
    v2f a = *(const v2f*)(Mr + kk);
    v2f b = *(const v2f*)(vec + kk);
    acc = wmma4(a, b, acc);
  }
  if (m == 0) {
    int ob = rbase + half * 8;
    for (int v = 0; v < 8; ++v) {
      int r = ob + v;
      if (r < rows) out[r] = acc[v];
    }
  }
}

// ---------- LN over magnitude + phase reconstruction + ReLU modulation ----------
__global__ void k_ln_phase(const float* ln_scale, const float* ln_shift, const float* mod_b, float* ws) {
  __shared__ float sbuf[1024];
  int j = threadIdx.x;
  float r = ws[OFF_RR + j];
  float i_ = ws[OFF_II + j];
  float hyp = sqrtf(r * r + i_ * i_);
  float mag = hyp + EPSF;
  float ssum = block_reduce_sum(mag, sbuf);
  float ssq = block_reduce_sum(mag * mag, sbuf);
  float mu = ssum / (float)DD;
  float var = (ssq - (float)DD * mu * mu) / (float)(DD - 1);  // ddof=1
  float magn = (mag - mu) * rsqrtf(var + EPSF) * ln_scale[j] + ln_shift[j];
  float ca = (hyp > 0.0f) ? (r / hyp) : 1.0f;   // cos(atan2(ii,rr))
  float sa = (hyp > 0.0f) ? (i_ / hyp) : 0.0f;  // sin(atan2(ii,rr))
  float zrv = magn * ca;
  float ziv = magn * sa;
  float mag2 = sqrtf(zrv * zrv + ziv * ziv) + EPSF;
  float s = fmaxf(mag2 + mod_b[j], 0.0f) / mag2;
  ws[OFF_ZR + j] = zrv * s;
  ws[OFF_ZI + j] = ziv * s;
}

// ---------- halt gate + ctrl softmax ----------
__global__ void k_gates(const float* halt_w, const float* halt_b, const float* ctrl_w,
                        const float* ctrl_b, float* ws) {
  __shared__ float sh[4][256];
  int t = threadIdx.x;
  float h = 0.f, c0 = 0.f, c1 = 0.f, c2 = 0.f;
  for (int k = t; k < DD2; k += 256) {
    float f = (k < DD) ? ws[OFF_ZR + k] : ws[OFF_ZI + k - DD];
    h  += f * halt_w[k];
    c0 += f * ctrl_w[k];
    c1 += f * ctrl_w[DD2 + k];
    c2 += f * ctrl_w[2 * DD2 + k];
  }
  sh[0][t] = h; sh[1][t] = c0; sh[2][t] = c1; sh[3][t] = c2;
  __syncthreads();
  for (int s = 128; s > 0; s >>= 1) {
    if (t < s)
      for (int q = 0; q < 4; ++q) sh[q][t] += sh[q][t + s];
    __syncthreads();
  }
  if (t == 0) {
    float p = 1.0f / (1.0f + expf(-(sh[0][0] + halt_b[0])));
    float a0 = sh[1][0] + ctrl_b[0];
    float a1 = sh[2][0] + ctrl_b[1];
    float a2 = sh[3][0] + ctrl_b[2];
    float mx = fmaxf(a0, fmaxf(a1, a2));
    float e0 = expf(a0 - mx), e1 = expf(a1 - mx), e2 = expf(a2 - mx);
    float es = e0 + e1 + e2;
    ws[OFF_SCAL + 0] = p;
    ws[OFF_SCAL + 1] = e0 / es;  // push
    ws[OFF_SCAL + 2] = e1 / es;  // pop
    ws[OFF_SCAL + 3] = e2 / es;  // noop
  }
}

// ---------- differentiable stack update + read; build zf ----------
__global__ void k_stack(float* ws) {
  __shared__ float up[STACKN], np[STACKN], snorm;
  int j = threadIdx.x;
  float push = ws[OFF_SCAL + 1], pop = ws[OFF_SCAL + 2], noop = ws[OFF_SCAL + 3];
  if (j < STACKN) {
    float u = ws[OFF_PTR + ((j + STACKN - 1) & (STACKN - 1))];
    float d = ws[OFF_PTR + ((j + 1) & (STACKN - 1))];
    float c = ws[OFF_PTR + j];
    up[j] = u;
    np[j] = push * u + pop * d + noop * c;
  }
  __syncthreads();
  if (j == 0) {
    float s = 0.f;
    for (int m2 = 0; m2 < STACKN; ++m2) s += np[m2];
    snorm = s + EPSF;
  }
  __syncthreads();
  if (j < STACKN) {
    np[j] = np[j] / snorm;
    ws[OFF_PTR + j] = np[j];
  }
  __syncthreads();
  float z = ws[OFF_ZR + j];
  float rd = 0.f;
  for (int m2 = 0; m2 < STACKN; ++m2) {
    float mv = push * up[m2] * z + (1.0f - push) * ws[OFF_MEM + m2 * DD + j];
    ws[OFF_MEM + m2 * DD + j] = mv;
    rd += mv * np[m2];
  }
  float zn = z + rd;
  ws[OFF_ZR + j] = zn;
  ws[OFF_ZF + j] = zn;
  ws[OFF_ZF + DD + j] = ws[OFF_ZI + j];
}

// ---------- argmin over codebook distances (norm - 2*dot - GBS*sigmoid(adj[prev])) ----------
__global__ void k_argmin(const float* adj, int have_prev, float* ws) {
  __shared__ float bv[1024];
  __shared__ int bidx[1024];
  int t = threadIdx.x;
  int* symp = (int*)(ws + OFF_SYM);
  int prev = have_prev ? symp[0] : 0;
  float best = 3.4e38f;
  int bi = 0;
  for (int i = t; i < NSYM; i += 1024) {
    float d = ws[OFF_CBN + i] - 2.0f * ws[OFF_DOTS + i];
    if (have_prev) d -= GBSF * (1.0f / (1.0f + expf(-adj[(long)prev * NSYM + i])));
    if (d < best || (d == best && i < bi)) { best = d; bi = i; }
  }
  bv[t] = best; bidx[t] = bi;
  __syncthreads();
  for (int s = 512; s > 0; s >>= 1) {
    if (t < s) {
      float v2 = bv[t + s]; int i2 = bidx[t + s];
      if (v2 < bv[t] || (v2 == bv[t] && i2 < bidx[t])) { bv[t] = v2; bidx[t] = i2; }
    }
    __syncthreads();
  }
  if (t == 0) symp[0] = bidx[0];
}

// ---------- VQ blend + vq_loss + ACT accumulation ----------
__global__ void k_update(const float* codebook, int is_last, float* ws) {
  __shared__ float sbuf[1024];
  int j = threadIdx.x;
  int s = ((int*)(ws + OFF_SYM))[0];
  float halt = ws[OFF_SCAL + 4];
  float remain = ws[OFF_SCAL + 5];
  float p = ws[OFF_SCAL + 0];
  float running = (halt < ACT_TF) ? 1.0f : 0.0f;
  float pt = is_last ? remain : p * running;
  float zqr = codebook[(long)s * DD2 + j];
  float zqi = codebook[(long)s * DD2 + DD + j];
  float zrv = ws[OFF_ZR + j];
  float ziv = ws[OFF_ZI + j];
  float dr = zqr - zrv, di = zqi - ziv;
  float dsum = block_reduce_sum(dr * dr + di * di, sbuf);
  float zrn = 0.7f * zrv + 0.3f * zqr;
  float zin = 0.7f * ziv + 0.3f * zqi;
  ws[OFF_ZR + j] = zrn;
  ws[OFF_ZI + j] = zin;
  ws[OFF_ZRACC + j] += pt * zrn;
  ws[OFF_ZIACC + j] += pt * zin;
  if (j == 0) {
    ws[OFF_SCAL + 7] += (1.0f + CCF) * dsum / (float)DD2;  // vq_loss
    ws[OFF_SCAL + 4] = halt + pt;
    ws[OFF_SCAL + 5] = remain - pt;
    ws[OFF_SCAL + 6] += running;  // ponder
  }
}

// ---------- build final zf = concat(zr_acc, zi_acc) ----------
__global__ void k_final_zf(float* ws) {
  int j = blockIdx.x * blockDim.x + threadIdx.x;
  if (j < DD2) ws[OFF_ZF + j] = (j < DD) ? ws[OFF_ZRACC + j] : ws[OFF_ZIACC + j - DD];
}

// ---------- epilogue: logits + bias, ponder, vq_loss -> d_out ----------
__global__ void k_epilogue(const float* out_b, const float* ws, float* out) {
  int i = blockIdx.x * blockDim.x + threadIdx.x;
  if (i < VOCAB) out[i] = ws[OFF_DOTS + i] + out_b[i];
  else if (i == VOCAB) out[i] = ws[OFF_SCAL + 6];
  else if (i == VOCAB + 1) out[i] = ws[OFF_SCAL + 7];
}

extern "C" void kernel_launch(void* const* d_in, const int* in_sizes, int n_in,
                              void* d_out, int out_size, void* d_ws, size_t ws_size,
                              hipStream_t stream) {
  const int*   x         = (const int*)d_in[0];
  const float* mag_emb   = (const float*)d_in[1];
  const float* phase_emb = (const float*)d_in[2];
  const float* Wr        = (const float*)d_in[3];
  const float* Wi        = (const float*)d_in[4];
  const float* ln_scale  = (const float*)d_in[5];
  const float* ln_shift  = (const float*)d_in[6];
  const float* mod_b     = (const float*)d_in[7];
  const float* halt_w    = (const float*)d_in[8];
  const float* halt_b    = (const float*)d_in[9];
  const float* ctrl_w    = (const float*)d_in[10];
  const float* ctrl_b    = (const float*)d_in[11];
  const float* codebook  = (const float*)d_in[12];
  const float* adj       = (const float*)d_in[13];
  const float* out_w     = (const float*)d_in[14];
  const float* out_b     = (const float*)d_in[15];
  float* ws  = (float*)d_ws;
  float* out = (float*)d_out;

  k_init<<<64, 1024, 0, stream>>>(x, mag_emb, phase_emb, ws);
  k_cbnorm<<<NSYM, 256, 0, stream>>>(codebook, ws);

  for (int t = 0; t < 8; ++t) {
    k_cplx_mv<<<DD / 16, 32, 0, stream>>>(Wr, Wi, ws, ws);
    k_ln_phase<<<1, 1024, 0, stream>>>(ln_scale, ln_shift, mod_b, ws);
    k_gates<<<1, 256, 0, stream>>>(halt_w, halt_b, ctrl_w, ctrl_b, ws);
    k_stack<<<1, 1024, 0, stream>>>(ws);
    k_mv_dot<<<NSYM / 16, 32, 0, stream>>>(codebook, ws + OFF_ZF, ws + OFF_DOTS, NSYM, DD2);
    k_argmin<<<1, 1024, 0, stream>>>(adj, t > 0 ? 1 : 0, ws);
    k_update<<<1, 1024, 0, stream>>>(codebook, t == 7 ? 1 : 0, ws);
  }

  k_final_zf<<<2, 1024, 0, stream>>>(ws);
  k_mv_dot<<<(VOCAB + 15) / 16, 32, 0, stream>>>(out_w, ws + OFF_ZF, ws + OFF_DOTS, VOCAB, DD2);
  k_epilogue<<<(VOCAB + 2 + 255) / 256, 256, 0, stream>>>(out_b, ws, out);
}